// XCY_44375602102949
// MI455X (gfx1250) — compile-verified
//
#include <hip/hip_runtime.h>
#include <hip/hip_bf16.h>
#include <math.h>

typedef __attribute__((ext_vector_type(2))) float v2f;
typedef __attribute__((ext_vector_type(8))) float v8f;

#define BATCH 8
#define CH 128
#define T1 4096

// ---- CDNA5 async global->LDS support (guarded; falls back to ld+ds_store) --
// Builtin signature (from hipcc diagnostic): (v4i* AS1, v4i* AS3, imm, imm).
#if defined(__gfx1250__) && __has_builtin(__builtin_amdgcn_global_load_async_to_lds_b128)
#define HAVE_ASYNC_LDS 1
typedef int v4i_vs __attribute__((vector_size(16)));
typedef __attribute__((address_space(1))) v4i_vs as1_v4i;
typedef __attribute__((address_space(3))) v4i_vs as3_v4i;
#if __has_builtin(__builtin_amdgcn_s_wait_asynccnt)
#define ASYNC_WAIT(n) __builtin_amdgcn_s_wait_asynccnt(n)
#else
#define ASYNC_WAIT(n) asm volatile("s_wait_asynccnt %0" ::"i"(n))
#endif
#else
#define HAVE_ASYNC_LDS 0
#define ASYNC_WAIT(n) ((void)0)
#endif

// ---- Kernel 1a: tiled transpose (B,C,T) -> (B,T,C), coalesced both sides --
__global__ __launch_bounds__(256)
void k_transpose(const float* __restrict__ x, float* __restrict__ xf)
{
    __shared__ float tile[32][33];
    const int b = blockIdx.z;
    const int tbase = blockIdx.x * 32;   // token tile
    const int cbase = blockIdx.y * 32;   // channel tile
    const int tx = threadIdx.x & 31;
    const int ty = threadIdx.x >> 5;     // 0..7
    #pragma unroll
    for (int r = ty; r < 32; r += 8)
        tile[r][tx] = x[((size_t)(b * CH + cbase + r)) * T1 + tbase + tx];
    __syncthreads();
    #pragma unroll
    for (int r = ty; r < 32; r += 8)
        xf[((size_t)b * T1 + tbase + r) * CH + cbase + tx] = tile[tx][r];
}

// ---- Kernel 1b: per-token inverse norm over (B,T,C), coalesced -----------
__global__ __launch_bounds__(128)
void k_row_norm(const float* __restrict__ xf, float* __restrict__ inv_norm, int T)
{
    __shared__ float red[128];
    const int t = blockIdx.x;
    const int b = blockIdx.y;
    const int c = threadIdx.x;
    float v = xf[((size_t)b * T + t) * CH + c];
    red[c] = v * v;
    __syncthreads();
    for (int s = 64; s > 0; s >>= 1) {
        if (c < s) red[c] += red[c + s];
        __syncthreads();
    }
    if (c == 0) inv_norm[(size_t)b * T + t] = rsqrtf(red[0]);
}

// ---- Kernel 2: sim GEMM (fp32 WMMA) + distance bias + row argmax ----------
// Two waves per block share a double-buffered LDS B-panel staged with async
// global->LDS (ASYNCcnt) copies. Each wave owns one 16-row A tile.
#define PANEL_STRIDE 132            // floats; pad 128->132 for bank-conflict-free ds_load_b64
__global__ __launch_bounds__(64)
void k_scores_argmax(const float* __restrict__ src,      // (B, T, C)
                     const float* __restrict__ inv_norm, // (B, T)
                     int* __restrict__ dst,              // (B, h)
                     const float* __restrict__ sim_w_p,
                     const float* __restrict__ dist_w_p,
                     int T, int h, int width)
{
    __shared__ float panel[2][16 * PANEL_STRIDE];

    const int tid  = threadIdx.x;          // 0..63
    const int wave = tid >> 5;             // 0,1
    const int lane = tid & 31;
    const int half = lane >> 4;
    const int l16  = lane & 15;
    const int itile = blockIdx.x * 2 + wave;
    const int b     = blockIdx.y;

    const float sim_w  = *sim_w_p;
    const float dist_w = *dist_w_p;

    const float* srcb = src + (size_t)b * T * CH;
    const float* invb = inv_norm + (size_t)b * T;

    // A fragments for K=128 (32 steps of K=4). Lane: row M=l16, K=2*half+{0,1}.
    const int arow = itile * 16 + l16;
    const float* aptr = srcb + (size_t)arow * CH + half * 2;
    v2f afrag[32];
    #pragma unroll
    for (int k = 0; k < 32; ++k) afrag[k] = *(const v2f*)(aptr + k * 4);

    // Per-lane C-slot metadata: slot v <-> row M = half*8 + v, col N = l16.
    float inv_na[8], arf[8], acf[8];
    #pragma unroll
    for (int v = 0; v < 8; ++v) {
        int i = itile * 16 + half * 8 + v;
        inv_na[v] = invb[i];
        arf[v] = (float)(i / width);
        acf[v] = (float)(i % width);
    }

    float bestV[8];
    int   bestJ[8];
    #pragma unroll
    for (int v = 0; v < 8; ++v) { bestV[v] = -3.402823e38f; bestJ[v] = 0; }

    // Stage one 16x128 B panel (8KB contiguous) into LDS buffer `buf`.
    auto stage = [&](int jt, int buf) {
        const float* gbase = srcb + (size_t)(h + jt * 16) * CH;
        #pragma unroll
        for (int q = 0; q < 8; ++q) {
            int f4  = q * 64 + tid;        // float4 index in panel (0..511)
            int row = f4 >> 5;             // 32 float4 per row
            int c4  = f4 & 31;
            const float* gp = gbase + f4 * 4;
            float* lp = &panel[buf][row * PANEL_STRIDE + c4 * 4];
#if HAVE_ASYNC_LDS
            __builtin_amdgcn_global_load_async_to_lds_b128(
                (as1_v4i*)gp, (as3_v4i*)lp, 0, 0);
#else
            *(float4*)lp = *(const float4*)gp;
#endif
        }
    };

    const int jtiles = h >> 4;
    stage(0, 0);
    for (int jt = 0; jt < jtiles; ++jt) {
        const int cur = jt & 1;
        if (jt + 1 < jtiles) {
            stage(jt + 1, cur ^ 1);
            ASYNC_WAIT(8);                 // this wave's `cur` panel complete
        } else {
            ASYNC_WAIT(0);
        }
        __syncthreads();                    // partner's panel half complete too

        // B fragments from LDS: lane -> column N=l16, K=2*half+{0,1} per step.
        const float* prow = &panel[cur][l16 * PANEL_STRIDE + half * 2];
        v2f bfrag[32];
        #pragma unroll
        for (int k = 0; k < 32; ++k) bfrag[k] = *(const v2f*)(prow + k * 4);

        v8f acc = {};
        #pragma unroll
        for (int k = 0; k < 32; ++k)
            acc = __builtin_amdgcn_wmma_f32_16x16x4_f32(false, afrag[k], false, bfrag[k],
                                                        (short)0, acc, false, false);

        const int   jcol   = jt * 16 + l16;       // column token (b-half local)
        const int   g      = h + jcol;            // global token index
        const float inv_nb = invb[g];
        const float br     = (float)(g / width);
        const float bc     = (float)(g % width);
        #pragma unroll
        for (int v = 0; v < 8; ++v) {
            float dr   = arf[v] - br;
            float dc   = acf[v] - bc;
            float dist = sqrtf(dr * dr + dc * dc);
            float s    = sim_w * (acc[v] * inv_na[v] * inv_nb) + dist_w / (dist + 1e-6f);
            if (s > bestV[v]) { bestV[v] = s; bestJ[v] = jcol; }  // j monotonic: first max kept
        }
        __syncthreads();                    // protect buffer recycle next iteration
    }

    // Reduce argmax across the 16 lanes of each half; first-index tie-break.
    #pragma unroll
    for (int v = 0; v < 8; ++v) {
        float bv = bestV[v];
        int   bj = bestJ[v];
        #pragma unroll
        for (int off = 1; off < 16; off <<= 1) {
            float ov = __shfl_xor(bv, off, 32);
            int   oj = __shfl_xor(bj, off, 32);
            if (ov > bv || (ov == bv && oj < bj)) { bv = ov; bj = oj; }
        }
        if (l16 == 0) dst[(size_t)b * h + itile * 16 + half * 8 + v] = bj;
    }
}

// ---- Kernel 3: merged = copy of b-half, cnt = 1 ---------------------------
__global__ void k_merge_init(const float* __restrict__ src, float* __restrict__ merged,
                             float* __restrict__ cnt, int h, int T)
{
    size_t idx = (size_t)blockIdx.x * blockDim.x + threadIdx.x;
    size_t n = (size_t)BATCH * h * CH;
    if (idx < n) {
        int b   = (int)(idx / ((size_t)h * CH));
        int rem = (int)(idx % ((size_t)h * CH));
        int j   = rem / CH;
        int c   = rem % CH;
        merged[idx] = src[((size_t)b * T + h + j) * CH + c];
    }
    if (idx < (size_t)BATCH * h) cnt[idx] = 1.0f;
}

// ---- Kernel 4: scatter-add a-rows into merged at dst ----------------------
__global__ __launch_bounds__(128)
void k_merge_scatter(const float* __restrict__ src, const int* __restrict__ dst,
                     float* __restrict__ merged, float* __restrict__ cnt, int h, int T)
{
    const int i = blockIdx.x;
    const int b = blockIdx.y;
    const int c = threadIdx.x;
    const int d = dst[(size_t)b * h + i];
    atomicAdd(&merged[((size_t)b * h + d) * CH + c], src[((size_t)b * T + i) * CH + c]);
    if (c == 0) atomicAdd(&cnt[(size_t)b * h + d], 1.0f);
}

// ---- Kernel 5: divide by counts; optionally emit inv-norms for next merge -
__global__ __launch_bounds__(128)
void k_merge_finalize(float* __restrict__ merged, const float* __restrict__ cnt,
                      float* __restrict__ inv_norm_out, int h)
{
    __shared__ float red[128];
    const int j = blockIdx.x;
    const int b = blockIdx.y;
    const int c = threadIdx.x;
    const size_t row = (size_t)b * h + j;
    const float inv_c = 1.0f / cnt[row];
    float v = merged[row * CH + c] * inv_c;
    merged[row * CH + c] = v;
    if (inv_norm_out) {
        red[c] = v * v;
        __syncthreads();
        for (int s = 64; s > 0; s >>= 1) {
            if (c < s) red[c] += red[c + s];
            __syncthreads();
        }
        if (c == 0) inv_norm_out[row] = rsqrtf(red[0]);
    }
}

// ---- Kernel 6: 1x1 conv as fp32 WMMA GEMM + BN + SiLU ---------------------
__global__ __launch_bounds__(32)
void k_conv_bn_silu(const float* __restrict__ xin,   // (B, P, C)
                    const float* __restrict__ wgt,   // (O=CH, C=CH)
                    const float* __restrict__ gamma, const float* __restrict__ beta,
                    const float* __restrict__ mean,  const float* __restrict__ var,
                    float* __restrict__ out,         // (B, O, P)
                    int P)
{
    const int ptile = blockIdx.x;
    const int otile = blockIdx.y;
    const int b     = blockIdx.z;
    const int lane  = threadIdx.x;
    const int half  = lane >> 4;
    const int l16   = lane & 15;

    const int orow = otile * 16 + l16;
    const float* aptr = wgt + (size_t)orow * CH + half * 2;
    const int p = ptile * 16 + l16;
    const float* bptr = xin + ((size_t)b * P + p) * CH + half * 2;

    // Batch all fragment loads first so the memory pipe stays full.
    v2f afrag[32], bfrag[32];
    #pragma unroll
    for (int k = 0; k < 32; ++k) afrag[k] = *(const v2f*)(aptr + k * 4);
    #pragma unroll
    for (int k = 0; k < 32; ++k) bfrag[k] = *(const v2f*)(bptr + k * 4);

    v8f acc = {};
    #pragma unroll
    for (int k = 0; k < 32; ++k)
        acc = __builtin_amdgcn_wmma_f32_16x16x4_f32(false, afrag[k], false, bfrag[k],
                                                    (short)0, acc, false, false);

    #pragma unroll
    for (int v = 0; v < 8; ++v) {
        const int o  = otile * 16 + half * 8 + v;
        const float sc = gamma[o] * rsqrtf(var[o] + 1e-5f);
        float y = (acc[v] - mean[o]) * sc + beta[o];
        out[((size_t)b * CH + o) * P + ptile * 16 + l16] = y / (1.0f + expf(-y));
    }
}

extern "C" void kernel_launch(void* const* d_in, const int* in_sizes, int n_in,
                              void* d_out, int out_size, void* d_ws, size_t ws_size,
                              hipStream_t stream)
{
    const float* x      = (const float*)d_in[0];
    const float* sim_w  = (const float*)d_in[1];
    const float* dist_w = (const float*)d_in[2];
    const float* conv_w = (const float*)d_in[3];
    const float* gamma  = (const float*)d_in[4];
    const float* beta   = (const float*)d_in[5];
    const float* mean   = (const float*)d_in[6];
    const float* var    = (const float*)d_in[7];
    float* out = (float*)d_out;
    (void)in_sizes; (void)n_in; (void)out_size; (void)ws_size;

    const int H1 = 2048, H2 = 1024;

    char* ws = (char*)d_ws;
    size_t off = 0;
    auto take = [&](size_t bytes) {
        char* p = ws + off;
        off = (off + bytes + 255) & ~(size_t)255;
        return p;
    };
    float* xf      = (float*)take((size_t)BATCH * T1 * CH * 4);   // 16 MB
    float* inv1    = (float*)take((size_t)BATCH * T1 * 4);
    int*   dst1    = (int*)  take((size_t)BATCH * H1 * 4);
    float* merged1 = (float*)take((size_t)BATCH * H1 * CH * 4);   // 8 MB
    float* cnt1    = (float*)take((size_t)BATCH * H1 * 4);
    float* inv2    = (float*)take((size_t)BATCH * H1 * 4);
    int*   dst2    = (int*)  take((size_t)BATCH * H2 * 4);
    float* merged2 = (float*)take((size_t)BATCH * H2 * CH * 4);   // 4 MB
    float* cnt2    = (float*)take((size_t)BATCH * H2 * 4);

    // 1. tokens (tiled transpose) + norms (coalesced)
    k_transpose<<<dim3(T1 / 32, CH / 32, BATCH), 256, 0, stream>>>(x, xf);
    k_row_norm<<<dim3(T1, BATCH), 128, 0, stream>>>(xf, inv1, T1);

    // 2. merge #1: WMMA scores + argmax (width = 64)
    k_scores_argmax<<<dim3(H1 / 32, BATCH), 64, 0, stream>>>(xf, inv1, dst1,
                                                             sim_w, dist_w, T1, H1, 64);
    {
        size_t n = (size_t)BATCH * H1 * CH;
        k_merge_init<<<(unsigned)((n + 255) / 256), 256, 0, stream>>>(xf, merged1, cnt1, H1, T1);
        k_merge_scatter<<<dim3(H1, BATCH), 128, 0, stream>>>(xf, dst1, merged1, cnt1, H1, T1);
        k_merge_finalize<<<dim3(H1, BATCH), 128, 0, stream>>>(merged1, cnt1, inv2, H1);
    }

    // 3. merge #2: width = int(sqrt(2048)) = 45
    k_scores_argmax<<<dim3(H2 / 32, BATCH), 64, 0, stream>>>(merged1, inv2, dst2,
                                                             sim_w, dist_w, H1, H2, 45);
    {
        size_t n = (size_t)BATCH * H2 * CH;
        k_merge_init<<<(unsigned)((n + 255) / 256), 256, 0, stream>>>(merged1, merged2, cnt2, H2, H1);
        k_merge_scatter<<<dim3(H2, BATCH), 128, 0, stream>>>(merged1, dst2, merged2, cnt2, H2, H1);
        k_merge_finalize<<<dim3(H2, BATCH), 128, 0, stream>>>(merged2, cnt2, nullptr, H2);
    }

    // 4. 1x1 conv + BN + SiLU (WMMA)
    k_conv_bn_silu<<<dim3(H2 / 16, CH / 16, BATCH), 32, 0, stream>>>(merged2, conv_w,
                                                                     gamma, beta, mean, var,
                                                                     out, H2);
}